// INV_LGN_DUAL_26603027431988
// MI455X (gfx1250) — compile-verified
//
#include <hip/hip_runtime.h>

// ---------------- problem constants (from reference) ----------------
#define N_USERS   100000
#define N_ITEMS   50000
#define N_NODES   150000            // N_USERS + N_ITEMS
#define EMB       64
#define N_EDGES   1000000           // 2 * N_INTER
#define BATCH     4096
#define KEEP_PROB 0.8f
#define DECAY     1e-4f
// INV_TAU == 1.0

#define NODE_ELEMS (N_NODES * EMB)  // 9,600,000 floats

typedef __attribute__((ext_vector_type(2))) float v2f;
typedef __attribute__((ext_vector_type(8))) float v8f;

__device__ __forceinline__ void atomAddF(float* p, float v) {
  (void)__hip_atomic_fetch_add(p, v, __ATOMIC_RELAXED, __HIP_MEMORY_SCOPE_AGENT);
}

// 256-thread block reduction (wave32): shfl within wave, LDS across 8 waves.
__device__ __forceinline__ float blockReduce256(float v) {
  __shared__ float s[8];
  #pragma unroll
  for (int m = 16; m >= 1; m >>= 1) v += __shfl_xor(v, m, 32);
  int lane = threadIdx.x & 31, w = threadIdx.x >> 5;
  if (lane == 0) s[w] = v;
  __syncthreads();
  if (w == 0) {
    v = (lane < 8) ? s[lane] : 0.0f;
    #pragma unroll
    for (int m = 4; m >= 1; m >>= 1) v += __shfl_xor(v, m, 32);
  }
  return v;  // valid on thread 0
}

// ---- 1) masked, rescaled edge values: v = ev * floor(drop + keep) / keep ----
__global__ void k_mask(const float* __restrict__ ev, const float* __restrict__ dr,
                       float* __restrict__ vals) {
  int i = blockIdx.x * blockDim.x + threadIdx.x;
  if (i < N_EDGES) vals[i] = ev[i] * floorf(dr[i] + KEEP_PROB) * (1.0f / KEEP_PROB);
}

// ---- 2) concat [ue; ie] into emb and acc (acc = layer-0 term) ----
__global__ void k_init(const float* __restrict__ ue, const float* __restrict__ ie,
                       float* __restrict__ emb, float* __restrict__ acc) {
  int i = blockIdx.x * blockDim.x + threadIdx.x;
  if (i >= NODE_ELEMS) return;
  float v = ((i >> 6) < N_USERS) ? ue[i] : ie[i - N_USERS * EMB];
  emb[i] = v;
  acc[i] = v;
}

// ---- 3) edge-parallel SpMM: 16 lanes per edge, float4 per lane.
// Whole matrix (38MB) is L2-resident on MI455X (192MB L2); atomics are
// native global_atomic_add_f32 resolved in L2.
__global__ void k_spmm(const float* __restrict__ vals, const int* __restrict__ row,
                       const int* __restrict__ col, const float* __restrict__ x,
                       float* __restrict__ y) {
  int idx = blockIdx.x * blockDim.x + threadIdx.x;
  int e = idx >> 4;
  if (e >= N_EDGES) return;
  int l = idx & 15;
  // stream edge lists ahead (gfx1250 global_prefetch_b8, speculative RT)
  int pe = e + 4096 < N_EDGES ? e + 4096 : N_EDGES - 1;
  __builtin_prefetch(col + pe, 0, 0);
  __builtin_prefetch(row + pe, 0, 0);
  __builtin_prefetch(vals + pe, 0, 0);
  float v = vals[e];
  int c = col[e], r = row[e];
  float4 xv = *((const float4*)(x + c * EMB) + l);
  float* yp = y + r * EMB + l * 4;
  atomAddF(yp + 0, v * xv.x);
  atomAddF(yp + 1, v * xv.y);
  atomAddF(yp + 2, v * xv.z);
  atomAddF(yp + 3, v * xv.w);
}

// ---- 4) acc += y ----
__global__ void k_axpy(float* __restrict__ acc, const float* __restrict__ y) {
  int i = blockIdx.x * blockDim.x + threadIdx.x;
  if (i < NODE_ELEMS) acc[i] += y[i];
}

// ---- 5) BPR loss via WMMA f32 16x16x4: one wave handles 16 batch rows.
// pos/neg scores = diag(U·Pᵀ), accumulated over K=64 in 16 WMMA steps.
// f32 A(16x4)/B(4x16) layouts are symmetric: lane l, comp j holds
// (row|col = l%16, K = 2*(l>>4)+j) -> identical float2 gather for A and B.
__device__ __forceinline__ float pick8(v8f c, int m) {
  switch (m) {
    case 0: return c[0]; case 1: return c[1]; case 2: return c[2]; case 3: return c[3];
    case 4: return c[4]; case 5: return c[5]; case 6: return c[6]; default: return c[7];
  }
}

__global__ void k_bpr(const float* __restrict__ acc, const int* __restrict__ users,
                      const int* __restrict__ pos, const int* __restrict__ neg,
                      float* __restrict__ out) {
  int g = blockIdx.x;            // 256 groups of 16 batch rows
  int l = threadIdx.x;           // 0..31 (wave32)
  int rsel = l & 15;
  int hi = l >> 4;               // K-half select
  int b = g * 16 + rsel;
  const float* up = acc + (size_t)users[b] * EMB;
  const float* pp = acc + (size_t)(N_USERS + pos[b]) * EMB;
  const float* np = acc + (size_t)(N_USERS + neg[b]) * EMB;
  v8f cp = {}, cn = {};
  #pragma unroll
  for (int kb = 0; kb < EMB; kb += 4) {
    int off = kb + 2 * hi;
    v2f a  = *(const v2f*)(up + off);
    v2f bp = *(const v2f*)(pp + off);
    v2f bn = *(const v2f*)(np + off);
    cp = __builtin_amdgcn_wmma_f32_16x16x4_f32(false, a, false, bp, (short)0, cp, false, false);
    cn = __builtin_amdgcn_wmma_f32_16x16x4_f32(false, a, false, bn, (short)0, cn, false, false);
  }
  // C/D layout: vgpr r, lanes 0-15 -> (M=r, N=lane); lanes 16-31 -> (M=r+8, N=lane-16).
  // Diagonal (i,i): i<8 -> vgpr i, lane i; i>=8 -> vgpr (lane-24), lanes 24..31.
  int m = (l < 8) ? l : ((l >= 24) ? (l - 24) : -1);
  float term = 0.0f;
  if (m >= 0) {
    float ps = pick8(cp, m), ns = pick8(cn, m);
    float x = (ps - ns) * (1.0f / 16.0f);      // both operands are acc/4
    float s = 1.0f / (1.0f + expf(-x));
    term = -logf(s + 1e-10f);
  }
  #pragma unroll
  for (int sh = 16; sh >= 1; sh >>= 1) term += __shfl_xor(term, sh, 32);
  if (l == 0) atomAddF(out + 0, term * (1.0f / (float)BATCH));
}

// ---- 6) L2 regularizer on RAW tables ----
__global__ void k_reg(const float* __restrict__ ue, const float* __restrict__ ie,
                      const int* __restrict__ us, const int* __restrict__ ps,
                      const int* __restrict__ ns, float* __restrict__ out) {
  int i = blockIdx.x * blockDim.x + threadIdx.x;  // BATCH*EMB
  float v = 0.0f;
  if (i < BATCH * EMB) {
    int b = i >> 6, d = i & 63;
    float a = ue[(size_t)us[b] * EMB + d];
    float p = ie[(size_t)ps[b] * EMB + d];
    float n = ie[(size_t)ns[b] * EMB + d];
    v = a * a + p * p + n * n;
  }
  v = blockReduce256(v);
  if (threadIdx.x == 0) atomAddF(out + 1, v * (0.5f * DECAY / (float)BATCH));
}

// ---- 7) invariance loss: mean((a0-a1)^2 / 16) split users/items ----
__global__ void k_inv(const float* __restrict__ a0, const float* __restrict__ a1,
                      float* __restrict__ out) {
  int i = blockIdx.x * blockDim.x + threadIdx.x;
  float v = 0.0f;
  if (i < NODE_ELEMS) {
    float d = (a0[i] - a1[i]) * 0.25f;          // /4 layer-mean each
    float w = (i < N_USERS * EMB) ? (1.0f / (float)(N_USERS * EMB))
                                  : (1.0f / (float)(N_ITEMS * EMB));
    v = d * d * w;                               // INV_TAU = 1
  }
  v = blockReduce256(v);
  if (threadIdx.x == 0) atomAddF(out + 2, v);
}

extern "C" void kernel_launch(void* const* d_in, const int* in_sizes, int n_in,
                              void* d_out, int out_size, void* d_ws, size_t ws_size,
                              hipStream_t stream) {
  const float* embed_user      = (const float*)d_in[0];
  const float* embed_item      = (const float*)d_in[1];
  const float* embed_user_dual = (const float*)d_in[2];
  const float* embed_item_dual = (const float*)d_in[3];
  const int*   edge_row        = (const int*)d_in[4];
  const int*   edge_col        = (const int*)d_in[5];
  const float* edge_val        = (const float*)d_in[6];
  const float* drop_u          = (const float*)d_in[7];
  const int*   users           = (const int*)d_in[8];
  const int*   pos_items       = (const int*)d_in[9];
  const int*   neg_items       = (const int*)d_in[10];
  float* out = (float*)d_out;

  // workspace carve: vals(1M) + embA + embB + acc0 + acc1 (4 x 9.6M) ~ 158MB
  float* vals = (float*)d_ws;
  float* embA = vals + N_EDGES;
  float* embB = embA + NODE_ELEMS;
  float* acc0 = embB + NODE_ELEMS;
  float* acc1 = acc0 + NODE_ELEMS;

  hipMemsetAsync(out, 0, 3 * sizeof(float), stream);

  const int TB = 256;
  const int gElems = (NODE_ELEMS + TB - 1) / TB;     // 37500
  const int gEdges = (N_EDGES + TB - 1) / TB;        // 3907
  const int gSpmm  = (N_EDGES * 16) / TB;            // 62500 (exact)
  const int gReg   = (BATCH * EMB + TB - 1) / TB;    // 1024

  for (int k = 0; k < 2; ++k) {
    // tables order: k=0 -> dual, k=1 -> primary (matches drop_u[k])
    const float* ue = (k == 0) ? embed_user_dual : embed_user;
    const float* ie = (k == 0) ? embed_item_dual : embed_item;
    float* acc = (k == 0) ? acc0 : acc1;

    k_mask<<<gEdges, TB, 0, stream>>>(edge_val, drop_u + (size_t)k * N_EDGES, vals);
    k_init<<<gElems, TB, 0, stream>>>(ue, ie, embA, acc);

    float* cur = embA;
    float* nxt = embB;
    for (int layer = 0; layer < 3; ++layer) {
      hipMemsetAsync(nxt, 0, (size_t)NODE_ELEMS * sizeof(float), stream);
      k_spmm<<<gSpmm, TB, 0, stream>>>(vals, edge_row, edge_col, cur, nxt);
      k_axpy<<<gElems, TB, 0, stream>>>(acc, nxt);
      float* t = cur; cur = nxt; nxt = t;
    }

    k_bpr<<<BATCH / 16, 32, 0, stream>>>(acc, users, pos_items, neg_items, out);
    k_reg<<<gReg, TB, 0, stream>>>(ue, ie, users, pos_items, neg_items, out);
  }

  k_inv<<<gElems, TB, 0, stream>>>(acc0, acc1, out);
}